// SigmoidBCELoss_74500502716952
// MI455X (gfx1250) — compile-verified
//
#include <hip/hip_runtime.h>
#include <hip/hip_bf16.h>

// word2vec negative-sampling BCE loss on gfx1250.
//   inputs:  x[B,64] f32, w[V,64] f32, tgt[B] i32, neg[B,K] i32
//   output:  scalar f32 loss
// Memory-bound gather problem (~30 MB @ 23.3 TB/s ~ 1.3us floor); compute done
// in full f32 via V_WMMA_F32_16X16X4_F32 (16 chained K-steps = 16x16x64 tile).

typedef float v2f __attribute__((ext_vector_type(2)));
typedef float v8f __attribute__((ext_vector_type(8)));

#define B_SZ   16384
#define D_DIM  64
#define K_NEG  5

__device__ __forceinline__ float softplus_stable(float x) {
    // softplus(x) = max(x,0) + log1p(exp(-|x|))
    return fmaxf(x, 0.0f) + log1pf(__expf(-fabsf(x)));
}

// One wave (32 lanes) handles 16 batch rows. For each of the 6 sample tiles
// (1 positive + 5 negatives) it gathers 16 weight rows and computes the full
// 16x16 cross-dot matrix with 16 chained v_wmma_f32_16x16x4_f32; the diagonal
// holds the 16 wanted logits.
__global__ __launch_bounds__(32)
void w2v_bce_loss_kernel(const float* __restrict__ x,    // [B, 64]
                         const float* __restrict__ w,    // [V, 64]
                         const int*   __restrict__ tgt,  // [B]
                         const int*   __restrict__ neg,  // [B, K]
                         float*       __restrict__ partial) // [gridDim.x]
{
    const int lane = threadIdx.x;      // 0..31
    const int m    = lane & 15;        // row within tile / N column
    const int h    = lane >> 4;        // half-wave selector (K sub-pair)
    const int b0   = blockIdx.x * 16;
    const int row  = b0 + m;

    // ---- Preload A operand (input rows), ISA 32-bit 16x4 A layout:
    // lanes 0-15 hold K=4kk..4kk+1, lanes 16-31 hold K=4kk+2..4kk+3, M = lane%16.
    const float* xrow = x + (size_t)row * D_DIM + 2 * h;
    v2f a[16];
    #pragma unroll
    for (int kk = 0; kk < 16; ++kk)
        a[kk] = *(const v2f*)(xrow + 4 * kk);

    float acc = 0.0f;

    #pragma unroll
    for (int s = 0; s < 1 + K_NEG; ++s) {
        const int idx = (s == 0) ? tgt[row] : neg[row * K_NEG + (s - 1)];
        const float* wrow = w + (size_t)idx * D_DIM + 2 * h;

        // ---- B operand: B[k][n] = W[n][4kk+k]; same per-lane pattern as A
        // (VGPR0 -> K even, VGPR1 -> K odd; lanes 16-31 take the upper K pair).
        v2f bm[16];
        #pragma unroll
        for (int kk = 0; kk < 16; ++kk)
            bm[kk] = *(const v2f*)(wrow + 4 * kk);

        v8f c = {0.f, 0.f, 0.f, 0.f, 0.f, 0.f, 0.f, 0.f};
        #pragma unroll
        for (int kk = 0; kk < 16; ++kk)
            c = __builtin_amdgcn_wmma_f32_16x16x4_f32(
                    /*neg_a=*/false, a[kk], /*neg_b=*/false, bm[kk],
                    /*c_mod=*/(short)0, c, /*reuse_a=*/false, /*reuse_b=*/false);

        // ---- Diagonal extraction (C layout: VGPR r, lane L -> M=r+8h, N=m).
        // Lane owns diag element iff r = m - 8h is in [0,8): lanes 0-7 & 24-31.
        const int r = m - 8 * h;
        float d = 0.0f;
        #pragma unroll
        for (int j = 0; j < 8; ++j)
            d = (j == r) ? c[j] : d;            // cndmask selects, EXEC untouched

        const bool owns = (r >= 0) && (r < 8);
        float term;
        if (s == 0)
            term = softplus_stable(-d) * (1.0f / (float)B_SZ);
        else
            term = softplus_stable(d) * (1.0f / ((float)B_SZ * (float)K_NEG));
        acc += owns ? term : 0.0f;
    }

    // ---- wave32 reduction
    #pragma unroll
    for (int off = 16; off >= 1; off >>= 1)
        acc += __shfl_xor(acc, off, 32);

    if (lane == 0)
        partial[blockIdx.x] = acc;
}

// Deterministic final reduction of the 1024 per-wave partials.
__global__ __launch_bounds__(256)
void reduce_partials_kernel(const float* __restrict__ partial,
                            float* __restrict__ out, int n)
{
    __shared__ float sm[256];
    const int t = threadIdx.x;
    float s = 0.0f;
    for (int i = t; i < n; i += 256)
        s += partial[i];
    sm[t] = s;
    __syncthreads();
    #pragma unroll
    for (int stride = 128; stride >= 1; stride >>= 1) {
        if (t < stride) sm[t] += sm[t + stride];
        __syncthreads();
    }
    if (t == 0) out[0] = sm[0];
}

extern "C" void kernel_launch(void* const* d_in, const int* in_sizes, int n_in,
                              void* d_out, int out_size, void* d_ws, size_t ws_size,
                              hipStream_t stream) {
    const float* x   = (const float*)d_in[0];   // input_embedding [B,64]
    const float* w   = (const float*)d_in[1];   // output_weights  [V,64]
    const int*   tgt = (const int*)d_in[2];     // target_index    [B]
    const int*   neg = (const int*)d_in[3];     // neg_indices     [B,K]
    float* out = (float*)d_out;
    float* partial = (float*)d_ws;              // 1024 floats of scratch

    const int nblocks = B_SZ / 16;              // 1024 waves
    w2v_bce_loss_kernel<<<nblocks, 32, 0, stream>>>(x, w, tgt, neg, partial);
    reduce_partials_kernel<<<1, 256, 0, stream>>>(partial, out, nblocks);
}